// TransferNet_68882685494029
// MI455X (gfx1250) — compile-verified
//
#include <hip/hip_runtime.h>
#include <hip/hip_bf16.h>

// ---------------------------------------------------------------------------
// TransferNet forward on MI455X (gfx1250).
// ~300 GFLOP of f32 GEMM (desc BiGRU xp + recurrent scans) vs <1 GB HBM
// traffic -> compute bound. All GEMMs are X @ W.T with f32 data, so we use
// V_WMMA_F32_16X16X4_F32 (exact f32 semantics, matrix pipe throughput).
//  - Large-M GEMMs (M%32==0): 2x2 register-blocked wave tile (32x32 C),
//    4 loads : 4 WMMAs per k-step, next-k prefetch to hide load latency.
//  - M=16 GEMMs (batch-16 rows): 16x32 wave tile with fused bias/act/mul
//    epilogue (qcls writes sigmoid(logit)*last_e straight to d_out).
// ---------------------------------------------------------------------------

typedef float v2f __attribute__((ext_vector_type(2)));
typedef float v8f __attribute__((ext_vector_type(8)));

#define WMMA_F32X4(A, B, C) \
  __builtin_amdgcn_wmma_f32_16x16x4_f32(false, (A), false, (B), (short)0, (C), false, false)

__device__ __forceinline__ float sigm(float x) { return 1.0f / (1.0f + __expf(-x)); }

__device__ __forceinline__ float waveSum(float v) {
#pragma unroll
  for (int o = 16; o > 0; o >>= 1) v += __shfl_xor(v, o, 32);
  return v;
}
__device__ __forceinline__ float waveMax(float v) {
#pragma unroll
  for (int o = 16; o > 0; o >>= 1) v = fmaxf(v, __shfl_xor(v, o, 32));
  return v;
}

// ---------------------------------------------------------------------------
// 2x2-blocked WMMA f32 GEMM: C[M,N] = A[M,K] @ W[N,K]^T + bias[N]
// Wave computes a 32x32 C tile: 2 A-frags x 2 B-frags -> 4 WMMAs per k-step,
// each fragment reused twice. Requires M%32==0, N%32==0, K%4==0.
// A-frag (16x4 f32): lane<16 -> K={0,1}, lane>=16 -> K={2,3}, M=lane&15.
// B-frag: B[k][n] = W[n][k] -> same float2 pattern read from W row n.
// C VGPR v holds M = v + 8*(lane>=16), N = lane&15.
// ---------------------------------------------------------------------------
__global__ void __launch_bounds__(256) k_wmma_gemm32(
    const float* __restrict__ A, int lda, const float* __restrict__ W, int ldw,
    const float* __restrict__ bias, float* __restrict__ C, int ldc,
    int Mtiles32, int NChunks, int K) {
  int wid = blockIdx.x * (blockDim.x >> 5) + (threadIdx.x >> 5);
  if (wid >= Mtiles32 * NChunks) return;  // wave-uniform: EXEC stays all-ones
  int mt = wid / NChunks;
  int nc = wid % NChunks;
  int lane = threadIdx.x & 31;
  int mrow = lane & 15;
  int hi = lane >> 4;
  int nbase = nc * 32;

  const float* a0p = A + (long)(mt * 32 + mrow) * lda + 2 * hi;
  const float* a1p = a0p + (long)16 * lda;
  const float* w0p = W + (long)(nbase + mrow) * ldw + 2 * hi;
  const float* w1p = w0p + (long)16 * ldw;

  v8f c00 = {}, c01 = {}, c10 = {}, c11 = {};

  v2f a0 = *(const v2f*)(a0p);
  v2f a1 = *(const v2f*)(a1p);
  v2f b0 = *(const v2f*)(w0p);
  v2f b1 = *(const v2f*)(w1p);
  for (int k = 4; k < K; k += 4) {
    // prefetch next k-step while 4 WMMAs run on current fragments
    v2f a0n = *(const v2f*)(a0p + k);
    v2f a1n = *(const v2f*)(a1p + k);
    v2f b0n = *(const v2f*)(w0p + k);
    v2f b1n = *(const v2f*)(w1p + k);
    c00 = WMMA_F32X4(a0, b0, c00);
    c01 = WMMA_F32X4(a0, b1, c01);
    c10 = WMMA_F32X4(a1, b0, c10);
    c11 = WMMA_F32X4(a1, b1, c11);
    a0 = a0n; a1 = a1n; b0 = b0n; b1 = b1n;
  }
  c00 = WMMA_F32X4(a0, b0, c00);
  c01 = WMMA_F32X4(a0, b1, c01);
  c10 = WMMA_F32X4(a1, b0, c10);
  c11 = WMMA_F32X4(a1, b1, c11);

  int n0 = nbase + mrow, n1 = n0 + 16;
  float bb0 = bias ? bias[n0] : 0.0f;
  float bb1 = bias ? bias[n1] : 0.0f;
#pragma unroll
  for (int v = 0; v < 8; ++v) {
    int m = mt * 32 + v + 8 * hi;
    long r0 = (long)m * ldc;
    long r1 = (long)(m + 16) * ldc;
    C[r0 + n0] = c00[v] + bb0;
    C[r0 + n1] = c01[v] + bb1;
    C[r1 + n0] = c10[v] + bb0;
    C[r1 + n1] = c11[v] + bb1;
  }
}

// ---------------------------------------------------------------------------
// 16-row WMMA f32 GEMM (M%16==0): C = act(A @ W.T + bias) (* mul)
// Used for batch-16 matmuls (q-scan gh, cq, qcls) where only one M tile
// exists. Fused epilogue: act 0=none, 1=tanh, 2=sigmoid; optional elementwise
// multiply (qcls: * last_e, writing the final output directly).
// ---------------------------------------------------------------------------
__global__ void __launch_bounds__(256) k_wmma_gemm16(
    const float* __restrict__ A, int lda, const float* __restrict__ W, int ldw,
    const float* __restrict__ bias, const float* __restrict__ mul,
    float* __restrict__ C, int ldc, int Mtiles, int NChunks, int K, int act) {
  int wid = blockIdx.x * (blockDim.x >> 5) + (threadIdx.x >> 5);
  if (wid >= Mtiles * NChunks) return;
  int mt = wid / NChunks;
  int nc = wid % NChunks;
  int lane = threadIdx.x & 31;
  int mrow = lane & 15;
  int hi = lane >> 4;
  int nbase = nc * 32;

  const float* arow = A + (long)(mt * 16 + mrow) * lda + 2 * hi;
  const float* w0p = W + (long)(nbase + mrow) * ldw + 2 * hi;
  const float* w1p = w0p + (long)16 * ldw;

  v8f acc0 = {}, acc1 = {};
  v2f a = *(const v2f*)(arow);
  v2f b0 = *(const v2f*)(w0p);
  v2f b1 = *(const v2f*)(w1p);
  for (int k = 4; k < K; k += 4) {
    v2f an = *(const v2f*)(arow + k);
    v2f b0n = *(const v2f*)(w0p + k);
    v2f b1n = *(const v2f*)(w1p + k);
    acc0 = WMMA_F32X4(a, b0, acc0);
    acc1 = WMMA_F32X4(a, b1, acc1);
    a = an; b0 = b0n; b1 = b1n;
  }
  acc0 = WMMA_F32X4(a, b0, acc0);
  acc1 = WMMA_F32X4(a, b1, acc1);

  int n0 = nbase + mrow, n1 = n0 + 16;
  float bb0 = bias ? bias[n0] : 0.0f;
  float bb1 = bias ? bias[n1] : 0.0f;
#pragma unroll
  for (int v = 0; v < 8; ++v) {
    int m = mt * 16 + v + 8 * hi;
    float x0 = acc0[v] + bb0;
    float x1 = acc1[v] + bb1;
    if (act == 1) {
      x0 = tanhf(x0);
      x1 = tanhf(x1);
    } else if (act == 2) {
      x0 = sigm(x0);
      x1 = sigm(x1);
    }
    long base = (long)m * ldc;
    if (mul) {
      x0 *= mul[base + n0];
      x1 *= mul[base + n1];
    }
    C[base + n0] = x0;
    C[base + n1] = x1;
  }
}

// Embedding gather (float4 per thread) + padding mask per token.
__global__ void k_gather(const int* __restrict__ tok, int ntok,
                         const float* __restrict__ emb, float* __restrict__ x,
                         float* __restrict__ mask) {
  long idx = (long)blockIdx.x * blockDim.x + threadIdx.x;
  if (idx >= (long)ntok * 64) return;
  int tr = (int)(idx >> 6);
  int g = (int)(idx & 63);
  int id = tok[tr];
  ((float4*)(x + (long)tr * 256))[g] = ((const float4*)(emb + (long)id * 256))[g];
  if (g == 0) mask[tr] = (id != 0) ? 1.0f : 0.0f;
}

__global__ void k_fill0(float* __restrict__ p, long n) {
  long i = (long)blockIdx.x * blockDim.x + threadIdx.x;
  if (i < n) p[i] = 0.0f;
}

// One GRU scan step (elementwise gates). gh = h@Wh.T+bh, xp = x@Wx.T+bx.
// h lives strided inside emb buffer (row*512 + dirofs), hs optional (masked).
__global__ void k_gru_gate(const float* __restrict__ xp, const float* __restrict__ gh,
                           const float* __restrict__ mask, float* __restrict__ emb,
                           float* __restrict__ hs, int R, int L, int t, int dirofs) {
  int idx = blockIdx.x * blockDim.x + threadIdx.x;
  if (idx >= R * 256) return;
  int row = idx >> 8, j = idx & 255;
  const float* xr = xp + ((long)row * L + t) * 768;
  const float* gr = gh + (long)row * 768;
  float r = sigm(xr[j] + gr[j]);
  float z = sigm(xr[256 + j] + gr[256 + j]);
  float n = tanhf(xr[512 + j] + r * gr[512 + j]);
  float* hp = emb + (long)row * 512 + dirofs + j;
  float h = *hp;
  float hn = (1.0f - z) * n + z * h;
  float m = mask[(long)row * L + t];
  float hnew = m * hn + (1.0f - m) * h;
  *hp = hnew;
  if (hs) hs[((long)row * L + t) * 512 + dirofs + j] = hnew * m;
}

// scores[b,l] = dot(cq[b,:], word_h[b,l,:]) — one wave per (b,l).
__global__ void k_scores(const float* __restrict__ cq, const float* __restrict__ wh,
                         float* __restrict__ sc) {
  int w = (blockIdx.x * blockDim.x + threadIdx.x) >> 5;
  int lane = threadIdx.x & 31;
  if (w >= 512) return;
  int b = w >> 5;
  const float* c = cq + (long)b * 512;
  const float* v = wh + (long)w * 512;
  float s = 0.0f;
  for (int h = lane; h < 512; h += 32) s += c[h] * v[h];
  s = waveSum(s);
  if (lane == 0) sc[w] = s;
}

// softmax over L=32 per batch row; one wave per b (launch <<<1,512>>>).
__global__ void k_softmax(const float* __restrict__ sc, float* __restrict__ dist) {
  int b = threadIdx.x >> 5, lane = threadIdx.x & 31;
  float s = sc[b * 32 + lane];
  float mx = waveMax(s);
  float e = __expf(s - mx);
  float sum = waveSum(e);
  dist[b * 32 + lane] = e / sum;
}

// cw[b,h] = (sum_l dist[b,l]*word_h[b,l,h]) * rel_W[h]
__global__ void k_ctx_cw(const float* __restrict__ dist, const float* __restrict__ wh,
                         const float* __restrict__ relW, float* __restrict__ cw) {
  int idx = blockIdx.x * blockDim.x + threadIdx.x;
  if (idx >= 16 * 512) return;
  int b = idx >> 9, h = idx & 511;
  float s = 0.0f;
#pragma unroll 4
  for (int l = 0; l < 32; ++l) s += dist[b * 32 + l] * wh[((long)b * 32 + l) * 512 + h];
  cw[idx] = s * relW[h];
}

// d_prob[row] = sigmoid(dot(cw[b,:], d_emb[row,:]) + rel_b), one wave per row.
__global__ void k_dlogit(const float* __restrict__ cw, const float* __restrict__ de,
                         const float* __restrict__ relb, float* __restrict__ dp) {
  int w = (blockIdx.x * blockDim.x + threadIdx.x) >> 5;
  int lane = threadIdx.x & 31;
  if (w >= 8000) return;
  int b = w / 500;
  const float* c = cw + (long)b * 512;
  const float* d = de + (long)w * 512;
  float s = 0.0f;
  for (int h = lane; h < 512; h += 32) s += c[h] * d[h];
  s = waveSum(s);
  if (lane == 0) dp[w] = sigm(s + relb[0]);
}

// scatter: e_dst[b, obj] += e_src[b, sub] * d_prob
__global__ void k_scatter(const int* __restrict__ pt, const float* __restrict__ es,
                          const float* __restrict__ dp, float* __restrict__ ed) {
  int idx = blockIdx.x * blockDim.x + threadIdx.x;
  if (idx >= 8000) return;
  int b = idx / 500;
  int s = pt[idx * 2 + 0];
  int o = pt[idx * 2 + 1];
  float v = es[(long)b * 100000 + s] * dp[idx];
  atomicAdd(&ed[(long)b * 100000 + o], v);
}

__global__ void k_norm(float* __restrict__ e, long n) {
  long i = (long)blockIdx.x * blockDim.x + threadIdx.x;
  if (i >= n) return;
  float v = e[i];
  e[i] = v / fmaxf(v, 1.0f);
}

extern "C" void kernel_launch(void* const* d_in, const int* in_sizes, int n_in,
                              void* d_out, int out_size, void* d_ws, size_t ws_size,
                              hipStream_t stream) {
  (void)in_sizes; (void)n_in; (void)out_size; (void)ws_size;
  const int B = 16, QL = 32, DL = 12, RSZ = 500, T = 2;
  const int DW = 256, DH = 512, HP = 256, NE = 100000, G = 3 * HP;  // G=768
  const int QT = B * QL;          // 512 question tokens
  const int DTOK = B * RSZ * DL;  // 96000 description tokens
  const int DR = B * RSZ;         // 8000 description rows

  const int* questions = (const int*)d_in[0];
  const float* e_s = (const float*)d_in[1];
  const int* pair = (const int*)d_in[2];
  const int* desc = (const int*)d_in[3];
  const float* word_emb = (const float*)d_in[4];
  const float* q_Wx = (const float*)d_in[5];
  const float* q_Wh = (const float*)d_in[6];
  const float* q_bx = (const float*)d_in[7];
  const float* q_bh = (const float*)d_in[8];
  const float* d_Wx = (const float*)d_in[9];
  const float* d_Wh = (const float*)d_in[10];
  const float* d_bx = (const float*)d_in[11];
  const float* d_bh = (const float*)d_in[12];
  const float* step_W = (const float*)d_in[13];
  const float* step_b = (const float*)d_in[14];
  const float* rel_W = (const float*)d_in[15];
  const float* rel_b = (const float*)d_in[16];
  const float* qcls_W = (const float*)d_in[17];
  const float* qcls_b = (const float*)d_in[18];
  float* out = (float*)d_out;

  float* ws = (float*)d_ws;
  size_t off = 0;
  auto alloc = [&](size_t n) { float* p = ws + off; off += n; return p; };
  float* x_q = alloc((size_t)QT * DW);
  float* mk_q = alloc(QT);
  float* xp_q0 = alloc((size_t)QT * G);
  float* xp_q1 = alloc((size_t)QT * G);
  float* wordh = alloc((size_t)QT * DH);
  float* q_emb = alloc((size_t)B * DH);
  float* gh_q = alloc((size_t)B * G);
  float* cq = alloc((size_t)B * DH);
  float* sc = alloc(QT);
  float* dist = alloc(QT);
  float* cw = alloc((size_t)B * DH);
  float* x_d = alloc((size_t)DTOK * DW);
  float* mk_d = alloc(DTOK);
  float* xp_d = alloc((size_t)DTOK * G);  // reused across dirs/t
  float* d_emb = alloc((size_t)DR * DH);
  float* gh_d = alloc((size_t)DR * G);
  float* dprob = alloc(DR);
  float* eA = alloc((size_t)B * NE);
  float* eB = alloc((size_t)B * NE);

  // Large-M GEMM (M%32==0): 32x32 wave tiles, bias-only epilogue.
  auto gemm32 = [&](const float* Ap, int lda, const float* Wp, int ldw,
                    const float* bp, float* Cp, int ldc, int M, int N, int K) {
    long tw = (long)(M / 32) * (N / 32);
    int blocks = (int)((tw + 7) / 8);
    k_wmma_gemm32<<<blocks, 256, 0, stream>>>(Ap, lda, Wp, ldw, bp, Cp, ldc,
                                              M / 32, N / 32, K);
  };
  // M=16-friendly GEMM with fused act/mul epilogue.
  auto gemm16 = [&](const float* Ap, int lda, const float* Wp, int ldw,
                    const float* bp, const float* mp, float* Cp, int ldc, int M,
                    int N, int K, int act) {
    long tw = (long)(M / 16) * (N / 32);
    int blocks = (int)((tw + 7) / 8);
    k_wmma_gemm16<<<blocks, 256, 0, stream>>>(Ap, lda, Wp, ldw, bp, mp, Cp, ldc,
                                              M / 16, N / 32, K, act);
  };
  auto fill0 = [&](float* p, long n) {
    k_fill0<<<(int)((n + 255) / 256), 256, 0, stream>>>(p, n);
  };

  // --- Question BiGRU -----------------------------------------------------
  k_gather<<<(QT * 64 + 255) / 256, 256, 0, stream>>>(questions, QT, word_emb,
                                                      x_q, mk_q);
  gemm32(x_q, DW, q_Wx, DW, q_bx, xp_q0, G, QT, G, DW);
  gemm32(x_q, DW, q_Wx + (size_t)G * DW, DW, q_bx + G, xp_q1, G, QT, G, DW);
  fill0(q_emb, (long)B * DH);
  for (int t = 0; t < QL; ++t) {  // forward scan (h = q_emb[:, 0:256])
    gemm16(q_emb, DH, q_Wh, DW, q_bh, nullptr, gh_q, G, B, G, HP, 0);
    k_gru_gate<<<(B * HP + 255) / 256, 256, 0, stream>>>(xp_q0, gh_q, mk_q,
                                                         q_emb, wordh, B, QL, t, 0);
  }
  for (int t = QL - 1; t >= 0; --t) {  // backward scan (h = q_emb[:, 256:512])
    gemm16(q_emb + HP, DH, q_Wh + (size_t)G * DW, DW, q_bh + G, nullptr, gh_q,
           G, B, G, HP, 0);
    k_gru_gate<<<(B * HP + 255) / 256, 256, 0, stream>>>(xp_q1, gh_q, mk_q,
                                                         q_emb, wordh, B, QL, t, HP);
  }

  // --- Reasoning steps ----------------------------------------------------
  const float* esrc = e_s;
  float* elast = eA;
  for (int t = 0; t < T; ++t) {
    gemm16(q_emb, DH, step_W + (size_t)t * DH * DH, DH, step_b + (size_t)t * DH,
           nullptr, cq, DH, B, DH, DH, 1);  // tanh
    k_scores<<<(QT * 32 + 255) / 256, 256, 0, stream>>>(cq, wordh, sc);
    k_softmax<<<1, 512, 0, stream>>>(sc, dist);
    k_ctx_cw<<<(B * DH + 255) / 256, 256, 0, stream>>>(dist, wordh, rel_W, cw);

    // Description BiGRU over 8000 rows x 12 tokens.
    k_gather<<<(int)(((long)DTOK * 64 + 255) / 256), 256, 0, stream>>>(
        desc + (size_t)t * DTOK, DTOK, word_emb, x_d, mk_d);
    fill0(d_emb, (long)DR * DH);
    for (int dir = 0; dir < 2; ++dir) {
      const float* Wx = d_Wx + (size_t)dir * G * DW;
      const float* Wh = d_Wh + (size_t)dir * G * DW;
      const float* bx = d_bx + dir * G;
      const float* bh = d_bh + dir * G;
      gemm32(x_d, DW, Wx, DW, bx, xp_d, G, DTOK, G, DW);  // dominant GEMM
      int ofs = dir * HP;
      for (int s = 0; s < DL; ++s) {
        int l = dir ? (DL - 1 - s) : s;
        gemm32(d_emb + ofs, DH, Wh, DW, bh, gh_d, G, DR, G, HP);
        k_gru_gate<<<(DR * HP + 255) / 256, 256, 0, stream>>>(
            xp_d, gh_d, mk_d, d_emb, nullptr, DR, DL, l, ofs);
      }
    }
    k_dlogit<<<(DR * 32 + 255) / 256, 256, 0, stream>>>(cw, d_emb, rel_b, dprob);

    float* edst = (t & 1) ? eB : eA;
    fill0(edst, (long)B * NE);
    k_scatter<<<(DR + 255) / 256, 256, 0, stream>>>(
        pair + (size_t)t * B * RSZ * 2, esrc, dprob, edst);
    k_norm<<<(int)(((long)B * NE + 255) / 256), 256, 0, stream>>>(edst, (long)B * NE);
    esrc = edst;
    elast = edst;
  }

  // --- Final: out = sigmoid(q_emb @ qcls_W.T + qcls_b) * last_e -----------
  // (HBM-bound: 205 MB of qcls_W at 23.3 TB/s ~ 9 us; fused epilogue avoids
  //  an extra 12.8 MB round trip for the elementwise multiply.)
  gemm16(q_emb, DH, qcls_W, DH, qcls_b, elast, out, NE, B, NE, DH, 2);
}